// ChamferCriterion_3264175145041
// MI455X (gfx1250) — compile-verified
//
#include <hip/hip_runtime.h>
#include <hip/hip_bf16.h>

// ---------------------------------------------------------------------------
// Chamfer loss on MI455X (gfx1250).
// Pairwise squared distances via V_WMMA_F32_16X16X4_F32 with the bias folded
// into the spare K slot:
//   A-role record: (-2x, -2y, -2z, 1)      (+BIG mask fill, pre-scaled)
//   B-role record: ( y0,  y1,  y2, |y|^2)  (-BIG mask fill)
//   => WMMA tile = |y_n|^2 - 2 * x_m . y_n   (add |x_m|^2 after the min)
// ---------------------------------------------------------------------------

typedef __attribute__((ext_vector_type(2))) float v2f;
typedef __attribute__((ext_vector_type(8))) float v8f;

namespace {
constexpr int   kB    = 32;
constexpr int   kN    = 2048;     // points per set (N == M)
constexpr float kBIG  = 1.0e4f;
constexpr float kBeta = 0.01f;
constexpr float kFMax = 3.402823466e38f;
}

// ---------------------------------------------------------------------------
// Prep: build the two record arrays per point set.
//   A-role (row operand):   (-2*c0, -2*c1, -2*c2, 1)   with +BIG fill
//   B-role (column operand): ( c0,   c1,   c2, norm)   with -BIG fill
// Note: norm of the -BIG fill equals norm of the +BIG fill (3*BIG^2), so the
// B-role .w field also serves as the row-norm |x|^2 table for the other pass.
// ---------------------------------------------------------------------------
__global__ __launch_bounds__(256) void prep_kernel(
    const float* __restrict__ outp, const unsigned char* __restrict__ omask,
    const float* __restrict__ tgt,  const unsigned char* __restrict__ tmask,
    float* __restrict__ outA, float* __restrict__ outB,
    float* __restrict__ tgtA, float* __restrict__ tgtB) {
  int i = blockIdx.x * blockDim.x + threadIdx.x;
  if (i >= kB * kN) return;

  {
    float x = outp[i * 3 + 0], y = outp[i * 3 + 1], z = outp[i * 3 + 2];
    bool m = omask[i] != 0;
    float ax = m ? kBIG : x, ay = m ? kBIG : y, az = m ? kBIG : z;   // +BIG fill
    float bx = m ? -kBIG : x, by = m ? -kBIG : y, bz = m ? -kBIG : z; // -BIG fill
    outA[i * 4 + 0] = -2.0f * ax; outA[i * 4 + 1] = -2.0f * ay;
    outA[i * 4 + 2] = -2.0f * az; outA[i * 4 + 3] = 1.0f;
    outB[i * 4 + 0] = bx; outB[i * 4 + 1] = by; outB[i * 4 + 2] = bz;
    outB[i * 4 + 3] = bx * bx + by * by + bz * bz;
  }
  {
    float x = tgt[i * 3 + 0], y = tgt[i * 3 + 1], z = tgt[i * 3 + 2];
    bool m = tmask[i] != 0;
    float ax = m ? kBIG : x, ay = m ? kBIG : y, az = m ? kBIG : z;
    float bx = m ? -kBIG : x, by = m ? -kBIG : y, bz = m ? -kBIG : z;
    tgtA[i * 4 + 0] = -2.0f * ax; tgtA[i * 4 + 1] = -2.0f * ay;
    tgtA[i * 4 + 2] = -2.0f * az; tgtA[i * 4 + 3] = 1.0f;
    tgtB[i * 4 + 0] = bx; tgtB[i * 4 + 1] = by; tgtB[i * 4 + 2] = bz;
    tgtB[i * 4 + 3] = bx * bx + by * by + bz * bz;
  }
}

// ---------------------------------------------------------------------------
// One Chamfer direction: rowmin[b][n] = xmask ? 0 : min_m |x_n - y_m|^2.
//
// Block = 256 threads = 8 waves; each wave owns one 16-row tile of one batch
// and sweeps all 128 16-column tiles, one v_wmma_f32_16x16x4_f32 each.
//
// f32 WMMA operand layout (ISA 7.12.2, 32-bit A 16x4 / B 4x16):
//   A: lanes 0-15 hold {K0,K1} of row M=lane; lanes 16-31 hold {K2,K3}.
//   B: lanes 0-15 hold {K0,K1} of col N=lane; lanes 16-31 hold {K2,K3}.
// With 16B records this is one float2 load per lane per operand:
//   lanes<16 read record comps (0,1); lanes>=16 read comps (2,3).
// Result: VGPR j, lanes 0-15 -> M=j, N=lane; lanes 16-31 -> M=j+8, N=lane-16.
//
// The WMMA already yields |y_n|^2 - 2 x.y, so the inner loop is just
// load -> wmma -> fmin; |x_m|^2 is added once after the lane reduction.
// ---------------------------------------------------------------------------
__global__ __launch_bounds__(256) void chamfer_min_pass(
    const float* __restrict__ xa,            // [B][N][4] A-role records of x
    const float* __restrict__ yb,            // [B][N][4] B-role records of y
    const float* __restrict__ xrec,          // [B][N][4] B-role records of x (.w = |x|^2)
    const unsigned char* __restrict__ xmask, // [B][N]
    float* __restrict__ rowmin) {            // [B][N]
  const int lane = threadIdx.x & 31;
  const int wave = threadIdx.x >> 5;
  const int l16  = lane & 15;
  const int half = lane >> 4;

  constexpr int kTilesPerBatch  = kN / 16;             // 128
  constexpr int kBlocksPerBatch = kTilesPerBatch / 8;  // 16

  const int b       = blockIdx.x / kBlocksPerBatch;
  const int rowTile = (blockIdx.x % kBlocksPerBatch) * 8 + wave;
  const int rowBase = rowTile * 16;

  const float* xab = xa + (size_t)b * kN * 4;
  const float* ybb = yb + (size_t)b * kN * 4;

  // A operand: held in registers for the whole column sweep.
  v2f a;
  {
    const float* p = xab + (size_t)(rowBase + l16) * 4 + half * 2;
    a.x = p[0];
    a.y = p[1];
  }

  float rmin[8];
#pragma unroll
  for (int j = 0; j < 8; ++j) rmin[j] = kFMax;

  for (int ct = 0; ct < kTilesPerBatch; ++ct) {
    const int n = ct * 16 + l16;
    const float* q = ybb + (size_t)n * 4 + half * 2;
    v2f bv;
    bv.x = q[0];
    bv.y = q[1];

    v8f c = {};
    c = __builtin_amdgcn_wmma_f32_16x16x4_f32(
        /*neg_a=*/false, a, /*neg_b=*/false, bv,
        /*c_mod=*/(short)0, c, /*reuse_a=*/false, /*reuse_b=*/false);

#pragma unroll
    for (int j = 0; j < 8; ++j) rmin[j] = fminf(rmin[j], c[j]);
  }

  // Min over the 16 lanes of each half-wave (columns of this lane group).
#pragma unroll
  for (int j = 0; j < 8; ++j) {
    float v = rmin[j];
#pragma unroll
    for (int m = 1; m < 16; m <<= 1) v = fminf(v, __shfl_xor(v, m, 32));
    rmin[j] = v;
  }

  // lane 0 -> rows rowBase+0..7, lane 16 -> rows rowBase+8..15
  if (l16 == 0) {
    const float* x2b = xrec + (size_t)b * kN * 4;
    const unsigned char* mb = xmask + (size_t)b * kN;
#pragma unroll
    for (int j = 0; j < 8; ++j) {
      const int r = rowBase + half * 8 + j;
      const float v = rmin[j] + x2b[(size_t)r * 4 + 3];  // + |x_r|^2
      rowmin[(size_t)b * kN + r] = mb[r] ? 0.0f : v;
    }
  }
}

// ---------------------------------------------------------------------------
// Per-batch: perBatch[b] = sum(min1[b])/nx[b] + sum(min2[b])/ny[b]
// ---------------------------------------------------------------------------
__global__ __launch_bounds__(256) void batch_reduce_kernel(
    const float* __restrict__ min1, const float* __restrict__ min2,
    const unsigned char* __restrict__ omask,
    const unsigned char* __restrict__ tmask,
    float* __restrict__ perBatch) {
  const int b = blockIdx.x;
  const int tid = threadIdx.x;

  float s1 = 0.0f, s2 = 0.0f;
  int c1 = 0, c2 = 0;
  for (int i = tid; i < kN; i += 256) {
    s1 += min1[(size_t)b * kN + i];
    s2 += min2[(size_t)b * kN + i];
    c1 += (omask[(size_t)b * kN + i] == 0);
    c2 += (tmask[(size_t)b * kN + i] == 0);
  }

  __shared__ float sh1[256], sh2[256];
  __shared__ int sc1[256], sc2[256];
  sh1[tid] = s1; sh2[tid] = s2; sc1[tid] = c1; sc2[tid] = c2;
  __syncthreads();
  for (int off = 128; off > 0; off >>= 1) {
    if (tid < off) {
      sh1[tid] += sh1[tid + off];
      sh2[tid] += sh2[tid + off];
      sc1[tid] += sc1[tid + off];
      sc2[tid] += sc2[tid + off];
    }
    __syncthreads();
  }
  if (tid == 0)
    perBatch[b] = sh1[0] / (float)sc1[0] + sh2[0] / (float)sc2[0];
}

// ---------------------------------------------------------------------------
// Finalize: 9 outputs = [loss, kl_loss, l2_loss, topdown_kl[5], beta]
// ---------------------------------------------------------------------------
__global__ void finalize_kernel(const float* __restrict__ perBatch,
                                const float* __restrict__ kls,
                                float* __restrict__ out) {
  if (threadIdx.x != 0 || blockIdx.x != 0) return;

  float l2 = 0.0f;
  for (int b = 0; b < kB; ++b) l2 += perBatch[b];
  l2 /= (float)kB;

  float kl = 0.0f;
  float col[5] = {0.0f, 0.0f, 0.0f, 0.0f, 0.0f};
  for (int b = 0; b < kB; ++b)
    for (int l = 0; l < 5; ++l) {
      const float v = kls[b * 5 + l];
      kl += v;
      col[l] += v;
    }
  kl /= (float)kB;

  const float scales[5] = {1.0f, 2.0f, 4.0f, 8.0f, 16.0f};
  out[0] = kBeta * kl + l2;
  out[1] = kl;
  out[2] = l2;
  for (int l = 0; l < 5; ++l) out[3 + l] = (col[l] / (float)kB) / (scales[l] * 16.0f);
  out[8] = kBeta;
}

// ---------------------------------------------------------------------------
extern "C" void kernel_launch(void* const* d_in, const int* in_sizes, int n_in,
                              void* d_out, int out_size, void* d_ws, size_t ws_size,
                              hipStream_t stream) {
  (void)in_sizes; (void)n_in; (void)out_size; (void)ws_size;

  const float*         out_set = (const float*)d_in[0];
  const unsigned char* omask   = (const unsigned char*)d_in[1];
  const float*         tgt_set = (const float*)d_in[2];
  const unsigned char* tmask   = (const unsigned char*)d_in[3];
  const float*         kls     = (const float*)d_in[4];
  float*               out     = (float*)d_out;

  // Workspace carve-up (floats).
  float* ws = (float*)d_ws;
  size_t off = 0;
  float* outA = ws + off; off += (size_t)kB * kN * 4;
  float* outB = ws + off; off += (size_t)kB * kN * 4;
  float* tgtA = ws + off; off += (size_t)kB * kN * 4;
  float* tgtB = ws + off; off += (size_t)kB * kN * 4;
  float* min1 = ws + off; off += (size_t)kB * kN;
  float* min2 = ws + off; off += (size_t)kB * kN;
  float* perBatch = ws + off; off += kB;

  // 1) Build WMMA-ready records (mask fill + pre-scale + norm folding).
  prep_kernel<<<(kB * kN + 255) / 256, 256, 0, stream>>>(
      out_set, omask, tgt_set, tmask, outA, outB, tgtA, tgtB);

  // 2) Two WMMA min passes (rows=output then rows=target).
  const int passBlocks = kB * (kN / 16) / 8;  // 512
  chamfer_min_pass<<<passBlocks, 256, 0, stream>>>(
      outA, tgtB, outB, omask, min1);
  chamfer_min_pass<<<passBlocks, 256, 0, stream>>>(
      tgtA, outB, tgtB, tmask, min2);

  // 3) Per-batch normalized sums.
  batch_reduce_kernel<<<kB, 256, 0, stream>>>(min1, min2, omask, tmask, perBatch);

  // 4) Scalars.
  finalize_kernel<<<1, 32, 0, stream>>>(perBatch, kls, out);
}